// BatchedModelManifoldGeodesicFlow_8624294331284
// MI455X (gfx1250) — compile-verified
//
#include <hip/hip_runtime.h>
#include <math.h>

typedef float v2f __attribute__((ext_vector_type(2)));
typedef float v8f __attribute__((ext_vector_type(8)));

#define NB 32    // batch
#define DD 128   // input dim
#define HH 256   // hidden dim
#define OO 10    // output dim

// ---------------------------------------------------------------------------
// Kernel 1: A = W2 * W2^T  (HH x HH),  B = W1^T * W1  (HH x HH)
// grid = HH blocks, 256 threads: thread n of block m computes A[m,n], B[m,n].
// ---------------------------------------------------------------------------
__global__ __launch_bounds__(256) void precompute_AB_kernel(
    const float* __restrict__ W1, const float* __restrict__ W2,
    float* __restrict__ Amat, float* __restrict__ Bmat)
{
  const int m = blockIdx.x;
  const int n = threadIdx.x;
  float a = 0.f;
#pragma unroll
  for (int o = 0; o < OO; ++o) a = fmaf(W2[m*OO + o], W2[n*OO + o], a);
  Amat[m*HH + n] = a;
  float bs = 0.f;
  for (int d = 0; d < DD; ++d) bs = fmaf(W1[d*HH + m], W1[d*HH + n], bs);
  Bmat[m*HH + n] = bs;
}

// ---------------------------------------------------------------------------
// Kernel 2: one workgroup per batch element.
//  C      = W1 * ( s .* ( A * (s' .* u^2) ) )                (geodesic term)
//  X      = diag(s') * A * diag(s)
//  Y      = X * B                (stored in LDS, 256 KB)
//  Z      = (B.*B) * X           (never stored; <Y,Z> accumulated on the fly)
//  ||dG||^2 = 2 * ( <Y,Z>_F + sum_{mp} B[m,p] Y[m,p] Y[p,m] )
//  a_i = -C_i / ((||dG||+1e-6) * (||v||+1e-6));  out = [v ; a - 0.1*dev]
// ---------------------------------------------------------------------------
__global__ __launch_bounds__(256) void geodesic_flow_kernel(
    const float* __restrict__ t,    const float* __restrict__ state,
    const float* __restrict__ x0g,  const float* __restrict__ x1g,
    const float* __restrict__ W1,   const float* __restrict__ b1,
    const float* __restrict__ Amat, const float* __restrict__ Bmat,
    float* __restrict__ out)
{
  extern __shared__ float Ysh[];                 // HH*HH floats (256 KB)
  __shared__ float xsh[DD], vsh[DD];
  __shared__ float ssh[HH], spsh[HH], qsh[HH], gsh[HH];
  __shared__ float red[256];

  const int b   = blockIdx.x;
  const int tid = threadIdx.x;

  const float tt     = t[0];
  const float window = 4.f * tt * (1.f - tt);

  // ---- interpolated point x, velocity v -------------------------------
  float devv = 0.f;
  if (tid < DD) {
    devv = state[b*DD + tid];
    const float a0 = x0g[b*DD + tid];
    const float a1 = x1g[b*DD + tid];
    xsh[tid] = a0 + tt*(a1 - a0) + window*devv;
    vsh[tid] = state[(NB + b)*DD + tid];
  }
  __syncthreads();

  // ---- z = x W1 + b1, u = v W1 (thread per hidden unit) ---------------
  float z = b1[tid], u = 0.f;
  for (int d = 0; d < DD; ++d) {
    const float w = W1[d*HH + tid];
    z = fmaf(xsh[d], w, z);
    u = fmaf(vsh[d], w, u);
  }
  const float h  = tanhf(z);
  const float s  = 1.f - h*h;       // sech^2
  const float sp = -2.f*h*s;        // d(sech^2)/dz
  ssh[tid]  = s;
  spsh[tid] = sp;
  qsh[tid]  = sp*u*u;
  __syncthreads();

  // ---- Aq = A * q  (A symmetric -> coalesced column read) -------------
  float aq = 0.f;
  for (int n = 0; n < HH; ++n) aq = fmaf(Amat[n*HH + tid], qsh[n], aq);
  gsh[tid] = s * aq;
  __syncthreads();

  // ---- C = W1 * g  (geodesic contraction, D-vector) -------------------
  float cv = 0.f;
  if (tid < DD) {
    for (int m = 0; m < HH; ++m) cv = fmaf(W1[tid*HH + m], gsh[m], cv);
  }

  // ---- ||v|| ----------------------------------------------------------
  red[tid] = (tid < DD) ? vsh[tid]*vsh[tid] : 0.f;
  __syncthreads();
  for (int off = 128; off > 0; off >>= 1) {
    if (tid < off) red[tid] += red[tid + off];
    __syncthreads();
  }
  const float vnorm = sqrtf(red[0]);
  __syncthreads();

  // ---- WMMA phase: lane decomposition for 16x16x4 f32 fragments -------
  const int wave = tid >> 5;       // 8 waves
  const int lane = tid & 31;
  const int lm   = lane & 15;      // M/N index within tile
  const int lh   = lane >> 4;      // K-half (A/B), M-half (C/D)
  const v8f vzero = {};

  // ============ Y = diag(sp) * A * diag(s) * B  -> LDS =================
  for (int mt = wave*2; mt < wave*2 + 2; ++mt) {        // 2 row-tiles/wave
    const int m = mt*16 + lm;
    for (int ptg = 0; ptg < 4; ++ptg) {                 // 4 groups of 4 col-tiles
      v8f acc[4];
#pragma unroll
      for (int j = 0; j < 4; ++j) acc[j] = vzero;
      for (int k = 0; k < 64; ++k) {                    // K = 256 in steps of 4
        const int n0 = k*4 + lh*2;
        v2f afrag;                                      // (A*diag(s))[m, n0:n0+2]
        afrag[0] = Amat[m*HH + n0    ] * ssh[n0    ];
        afrag[1] = Amat[m*HH + n0 + 1] * ssh[n0 + 1];
#pragma unroll
        for (int j = 0; j < 4; ++j) {
          const int p = (ptg*4 + j)*16 + lm;
          v2f bfrag;                                    // B[n0:n0+2, p]
          bfrag[0] = Bmat[ n0     *HH + p];
          bfrag[1] = Bmat[(n0 + 1)*HH + p];
          acc[j] = __builtin_amdgcn_wmma_f32_16x16x4_f32(
                       false, afrag, false, bfrag, (short)0, acc[j], false, false);
        }
      }
#pragma unroll
      for (int j = 0; j < 4; ++j) {
        const int pp = (ptg*4 + j)*16 + lm;
#pragma unroll
        for (int r = 0; r < 8; ++r) {                   // C/D: M = r + 8*lh
          const int mm = mt*16 + 8*lh + r;
          Ysh[mm*HH + pp] = spsh[mm] * acc[j][r];       // fold diag(sp) row scale
        }
      }
    }
  }
  __syncthreads();

  // ============ Z = (B.*B) * X tiles; accumulate <Y,Z>_F ===============
  float dotYZ = 0.f;
  for (int mt = wave*2; mt < wave*2 + 2; ++mt) {
    const int m = mt*16 + lm;
    for (int ptg = 0; ptg < 4; ++ptg) {
      v8f acc[4];
#pragma unroll
      for (int j = 0; j < 4; ++j) acc[j] = vzero;
      for (int k = 0; k < 64; ++k) {
        const int n0 = k*4 + lh*2;
        const float bb0 = Bmat[m*HH + n0];
        const float bb1 = Bmat[m*HH + n0 + 1];
        v2f afrag;                                      // (B.*B)[m, n0:n0+2]
        afrag[0] = bb0*bb0;
        afrag[1] = bb1*bb1;
        const float xr0 = spsh[n0];
        const float xr1 = spsh[n0 + 1];
#pragma unroll
        for (int j = 0; j < 4; ++j) {
          const int p  = (ptg*4 + j)*16 + lm;
          const float sc = ssh[p];
          v2f bfrag;                                    // X[n0:n0+2, p] = sp_n A[n,p] s_p
          bfrag[0] = xr0 * Amat[ n0     *HH + p] * sc;
          bfrag[1] = xr1 * Amat[(n0 + 1)*HH + p] * sc;
          acc[j] = __builtin_amdgcn_wmma_f32_16x16x4_f32(
                       false, afrag, false, bfrag, (short)0, acc[j], false, false);
        }
      }
#pragma unroll
      for (int j = 0; j < 4; ++j) {
        const int pp = (ptg*4 + j)*16 + lm;
#pragma unroll
        for (int r = 0; r < 8; ++r) {
          const int mm = mt*16 + 8*lh + r;
          dotYZ = fmaf(acc[j][r], Ysh[mm*HH + pp], dotYZ);
        }
      }
    }
  }

  // ---- cross term: sum_{mp} B[m,p] * Y[m,p] * Y[p,m] ------------------
  float cross = 0.f;
  for (int idx = tid; idx < HH*HH; idx += 256) {
    const int mm = idx >> 8;
    const int pp = idx & (HH - 1);
    cross = fmaf(Bmat[idx]*Ysh[idx], Ysh[pp*HH + mm], cross);
  }

  // ---- reduce, finalize ----------------------------------------------
  red[tid] = dotYZ + cross;
  __syncthreads();
  for (int off = 128; off > 0; off >>= 1) {
    if (tid < off) red[tid] += red[tid + off];
    __syncthreads();
  }
  const float normF = sqrtf(fmaxf(2.f * red[0], 0.f));

  if (tid < DD) {
    const float ai = -cv / ((normF + 1e-6f) * (vnorm + 1e-6f));
    out[b*DD + tid]        = vsh[tid];            // dev_velocity
    out[(NB + b)*DD + tid] = ai - 0.1f*devv;      // a + restoration
  }
}

// ---------------------------------------------------------------------------
extern "C" void kernel_launch(void* const* d_in, const int* in_sizes, int n_in,
                              void* d_out, int out_size, void* d_ws, size_t ws_size,
                              hipStream_t stream)
{
  (void)in_sizes; (void)n_in; (void)out_size; (void)ws_size;
  const float* t     = (const float*)d_in[0];
  const float* state = (const float*)d_in[1];
  const float* x0    = (const float*)d_in[2];
  const float* x1    = (const float*)d_in[3];
  const float* W1    = (const float*)d_in[4];
  const float* b1    = (const float*)d_in[5];
  const float* W2    = (const float*)d_in[6];
  // d_in[7] (b2) is mathematically irrelevant: only Jacobians of f appear.
  float* out  = (float*)d_out;
  float* Amat = (float*)d_ws;            // 256 KB
  float* Bmat = Amat + HH*HH;            // 256 KB

  precompute_AB_kernel<<<dim3(HH), dim3(256), 0, stream>>>(W1, W2, Amat, Bmat);
  geodesic_flow_kernel<<<dim3(NB), dim3(256), HH*HH*sizeof(float), stream>>>(
      t, state, x0, x1, W1, b1, Amat, Bmat, out);
}